// EdgeCentricRGCN_7275674600533
// MI455X (gfx1250) — compile-verified
//
#include <hip/hip_runtime.h>
#include <hip/hip_bf16.h>

typedef float v2f __attribute__((ext_vector_type(2)));
typedef float v8f __attribute__((ext_vector_type(8)));

#define HID 128
#define NGRAPH 64

// ---------------------------------------------------------------------------
// zero a buffer (float4 granularity)
// ---------------------------------------------------------------------------
__global__ __launch_bounds__(256) void zero_f4(float4* __restrict__ p, int n4) {
    int t = blockIdx.x * blockDim.x + threadIdx.x;
    if (t < n4) p[t] = make_float4(0.f, 0.f, 0.f, 0.f);
}

// ---------------------------------------------------------------------------
// copy buffer (float4 granularity): dst = src
// Used to seed agg = x, so GINE's (x + sum msg) needs no add in the GEMM.
// ---------------------------------------------------------------------------
__global__ __launch_bounds__(256) void copy_f4(const float4* __restrict__ s,
                                               float4* __restrict__ d, int n4) {
    int t = blockIdx.x * blockDim.x + threadIdx.x;
    if (t < n4) d[t] = s[t];
}

// ---------------------------------------------------------------------------
// x[i,h] = nw[i]*Wn[h] + bn[h]   (rank-1 node embedding)
// one thread per (node, 4 cols)
// ---------------------------------------------------------------------------
__global__ __launch_bounds__(256) void init_x_kernel(
    const float* __restrict__ nw, const float* __restrict__ Wn,
    const float* __restrict__ bn, float* __restrict__ x, int nThreads) {
    int tid = blockIdx.x * blockDim.x + threadIdx.x;
    if (tid >= nThreads) return;
    int i = tid >> 5;
    int c = (tid & 31) << 2;
    float w = nw[i];
    float4 wn = *(const float4*)(Wn + c);
    float4 b  = *(const float4*)(bn + c);
    float4 v;
    v.x = w * wn.x + b.x;
    v.y = w * wn.y + b.y;
    v.z = w * wn.z + b.z;
    v.w = w * wn.w + b.w;
    *(float4*)(x + (size_t)i * HID + c) = v;
}

// ---------------------------------------------------------------------------
// Edge message + scatter:  agg[dst] += relu(x[src] + ew*We + be)
// e is rank-1 -> recomputed in registers. One thread per (edge, 4 cols).
// x (25.6MB) and agg (25.6MB) both fit in the 192MB L2 -> gather/scatter
// stays on-die. unsafeAtomicAdd -> global_atomic_add_f32 (no CAS loop).
// ---------------------------------------------------------------------------
__global__ __launch_bounds__(256) void edge_scatter_kernel(
    const float* __restrict__ x, const long long* __restrict__ src,
    const long long* __restrict__ dst, const float* __restrict__ ew,
    const float* __restrict__ We, const float* __restrict__ be,
    float* __restrict__ agg, int nE) {
    int tid = blockIdx.x * blockDim.x + threadIdx.x;
    int e = tid >> 5;
    if (e >= nE) return;
    int c = (tid & 31) << 2;
    long long s = src[e];
    long long d = dst[e];
    float w = ew[e];
    float4 xv  = *(const float4*)(x + (size_t)s * HID + c);
    float4 wev = *(const float4*)(We + c);
    float4 bev = *(const float4*)(be + c);
    float4 m;
    m.x = xv.x + w * wev.x + bev.x; m.x = m.x > 0.f ? m.x : 0.f;
    m.y = xv.y + w * wev.y + bev.y; m.y = m.y > 0.f ? m.y : 0.f;
    m.z = xv.z + w * wev.z + bev.z; m.z = m.z > 0.f ? m.z : 0.f;
    m.w = xv.w + w * wev.w + bev.w; m.w = m.w > 0.f ? m.w : 0.f;
    float* ap = agg + (size_t)d * HID + c;
    unsafeAtomicAdd(ap + 0, m.x);
    unsafeAtomicAdd(ap + 1, m.y);
    unsafeAtomicAdd(ap + 2, m.z);
    unsafeAtomicAdd(ap + 3, m.w);
}

// ---------------------------------------------------------------------------
// Fused GEMM + bias + activation via V_WMMA_F32_16X16X4_F32 (fp32-exact).
//   Out[M,128] = act( A[M,128] @ W[128,128] + bias )
// Block = 256 threads = 8 waves; block handles a 16-row slab (50000 = 3125*16),
// wave w owns output columns [16w, 16w+16). K loop: 32 WMMAs of K=4, split
// across TWO accumulators so WMMA RAW chains interleave instead of serializing.
// A 16x4 f32 fragment (ISA 7.12.2): lanes 0-15 hold {K0,K1} of row M=lane,
// lanes 16-31 hold {K2,K3}; B 4x16 mirrors with lanes mapping to N.
// C/D: VGPR i -> rows M=i (lanes 0-15) / M=i+8 (lanes 16-31).
// ---------------------------------------------------------------------------
__global__ __launch_bounds__(256) void gemm_act_kernel(
    const float* __restrict__ A, const float* __restrict__ W,
    const float* __restrict__ bias, float* __restrict__ Out, int leaky) {
    const int lane = threadIdx.x & 31;
    const int wave = threadIdx.x >> 5;
    const int m0 = blockIdx.x * 16;
    const int n0 = wave * 16;
    const int l15 = lane & 15;
    const int hi = lane >> 4;          // 0: K pair {0,1}, 1: K pair {2,3}
    const int row = m0 + l15;          // A-matrix row owned by this lane
    const int col = n0 + l15;          // B/C/D column owned by this lane

    const float* arow = A + (size_t)row * HID;

    v8f acc0 = {0.f, 0.f, 0.f, 0.f, 0.f, 0.f, 0.f, 0.f};
    v8f acc1 = {0.f, 0.f, 0.f, 0.f, 0.f, 0.f, 0.f, 0.f};

#pragma unroll 4
    for (int k0 = 0; k0 < HID; k0 += 8) {
        // chunk 0: K = [k0, k0+4)
        {
            const int ka = k0 + (hi << 1);
            v2f a, b;
            a.x = arow[ka];
            a.y = arow[ka + 1];
            b.x = W[(size_t)ka * HID + col];
            b.y = W[(size_t)(ka + 1) * HID + col];
            acc0 = __builtin_amdgcn_wmma_f32_16x16x4_f32(
                false, a, false, b, (short)0, acc0, false, false);
        }
        // chunk 1: K = [k0+4, k0+8)  (independent accumulator -> overlaps)
        {
            const int ka = k0 + 4 + (hi << 1);
            v2f a, b;
            a.x = arow[ka];
            a.y = arow[ka + 1];
            b.x = W[(size_t)ka * HID + col];
            b.y = W[(size_t)(ka + 1) * HID + col];
            acc1 = __builtin_amdgcn_wmma_f32_16x16x4_f32(
                false, a, false, b, (short)0, acc1, false, false);
        }
    }

    const float bv = bias[col];
#pragma unroll
    for (int i = 0; i < 8; ++i) {
        const int r = m0 + i + (hi << 3);
        float v = acc0[i] + acc1[i] + bv;
        if (leaky) v = v > 0.f ? v : 0.01f * v;
        else       v = v > 0.f ? v : 0.f;
        Out[(size_t)r * HID + col] = v;
    }
}

// ---------------------------------------------------------------------------
// Global mean-pool accumulation: sums[batch[i]] += x[i]; one thread / (node,4c)
// ---------------------------------------------------------------------------
__global__ __launch_bounds__(256) void pool_scatter_kernel(
    const float* __restrict__ x, const long long* __restrict__ batch,
    float* __restrict__ sums, int nN) {
    int tid = blockIdx.x * blockDim.x + threadIdx.x;
    int i = tid >> 5;
    if (i >= nN) return;
    int c = (tid & 31) << 2;
    long long g = batch[i];
    float4 xv = *(const float4*)(x + (size_t)i * HID + c);
    float* sp = sums + (size_t)g * HID + c;
    unsafeAtomicAdd(sp + 0, xv.x);
    unsafeAtomicAdd(sp + 1, xv.y);
    unsafeAtomicAdd(sp + 2, xv.z);
    unsafeAtomicAdd(sp + 3, xv.w);
}

__global__ __launch_bounds__(256) void pool_count_kernel(
    const long long* __restrict__ batch, float* __restrict__ cnts, int nN) {
    int i = blockIdx.x * blockDim.x + threadIdx.x;
    if (i < nN) unsafeAtomicAdd(&cnts[batch[i]], 1.0f);
}

// ---------------------------------------------------------------------------
// Final MLP on [64,128] pooled features (negligible FLOPs -> single block).
//   h = leaky(pooled @ Wm1 + bm1);  out = sigmoid(h @ Wm2 + bm2)
// ---------------------------------------------------------------------------
__global__ __launch_bounds__(256) void mlp_kernel(
    const float* __restrict__ sums, const float* __restrict__ cnts,
    const float* __restrict__ Wm1, const float* __restrict__ bm1,
    const float* __restrict__ Wm2, const float* __restrict__ bm2,
    float* __restrict__ out) {
    __shared__ float H1[NGRAPH * HID];
    const int t = threadIdx.x;
    for (int i = t; i < NGRAPH * HID; i += 256) {
        const int g = i >> 7;
        const int o = i & (HID - 1);
        float c = cnts[g];
        c = c < 1.f ? 1.f : c;
        const float inv = 1.f / c;
        float acc = bm1[o];
        for (int k = 0; k < HID; ++k)
            acc += (sums[g * HID + k] * inv) * Wm1[k * HID + o];
        H1[i] = acc > 0.f ? acc : 0.01f * acc;
    }
    __syncthreads();
    if (t < NGRAPH) {
        float acc = bm2[0];
        for (int k = 0; k < HID; ++k)
            acc += H1[t * HID + k] * Wm2[k];
        out[t] = 1.f / (1.f + expf(-acc));
    }
}

// ---------------------------------------------------------------------------
extern "C" void kernel_launch(void* const* d_in, const int* in_sizes, int n_in,
                              void* d_out, int out_size, void* d_ws, size_t ws_size,
                              hipStream_t stream) {
    const float*     node_weight = (const float*)d_in[0];
    const float*     edge_weight = (const float*)d_in[1];
    const long long* edge_index  = (const long long*)d_in[2];
    const long long* batch       = (const long long*)d_in[3];
    const float* Wn  = (const float*)d_in[4];
    const float* bn  = (const float*)d_in[5];
    const float* We  = (const float*)d_in[6];
    const float* be  = (const float*)d_in[7];
    const float* W1a = (const float*)d_in[8];
    const float* b1a = (const float*)d_in[9];
    const float* W1b = (const float*)d_in[10];
    const float* b1b = (const float*)d_in[11];
    const float* W2a = (const float*)d_in[12];
    const float* b2a = (const float*)d_in[13];
    const float* W2b = (const float*)d_in[14];
    const float* b2b = (const float*)d_in[15];
    const float* Wm1 = (const float*)d_in[16];
    const float* bm1 = (const float*)d_in[17];
    const float* Wm2 = (const float*)d_in[18];
    const float* bm2 = (const float*)d_in[19];

    const int N = in_sizes[0];        // 50000
    const int E = in_sizes[1];        // 600000
    const long long* src = edge_index;
    const long long* dst = edge_index + E;

    // workspace layout (fp32)
    float* bufA = (float*)d_ws;                    // x        [N,128]
    float* bufB = bufA + (size_t)N * HID;          // agg      [N,128] (seeded = x)
    float* bufC = bufB + (size_t)N * HID;          // h        [N,128]
    float* sums = bufC + (size_t)N * HID;          // pooled   [64,128]
    float* cnts = sums + NGRAPH * HID;             // counts   [64]

    const int NH_T = N * 32;                       // thread count for (node,4col)
    const int EH_T = E * 32;                       // thread count for (edge,4col)
    const int B = 256;

    // x = nw*Wn + bn
    init_x_kernel<<<(NH_T + B - 1) / B, B, 0, stream>>>(node_weight, Wn, bn, bufA, NH_T);

    const float* Wa[2] = {W1a, W2a};
    const float* ba[2] = {b1a, b2a};
    const float* Wb[2] = {W1b, W2b};
    const float* bb[2] = {b1b, b2b};

    for (int layer = 0; layer < 2; ++layer) {
        // agg = x   (seed with self term -> GEMM reads one clean operand)
        copy_f4<<<(N * 32 + B - 1) / B, B, 0, stream>>>(
            (const float4*)bufA, (float4*)bufB, N * 32);
        // agg[dst] += relu(x[src] + ew*We + be)
        edge_scatter_kernel<<<(EH_T + B - 1) / B, B, 0, stream>>>(
            bufA, src, dst, edge_weight, We, be, bufB, E);
        // h = leaky_relu(agg @ Wa + ba)             [WMMA]
        gemm_act_kernel<<<N / 16, B, 0, stream>>>(bufB, Wa[layer], ba[layer], bufC, 1);
        // x = relu(h @ Wb + bb)                     [WMMA]
        gemm_act_kernel<<<N / 16, B, 0, stream>>>(bufC, Wb[layer], bb[layer], bufA, 0);
    }

    // pooled sums + counts (contiguous -> one zero pass)
    zero_f4<<<((NGRAPH * HID + NGRAPH) / 4 + B - 1) / B, B, 0, stream>>>(
        (float4*)sums, (NGRAPH * HID + NGRAPH) / 4);
    pool_scatter_kernel<<<(NH_T + B - 1) / B, B, 0, stream>>>(bufA, batch, sums, N);
    pool_count_kernel<<<(N + B - 1) / B, B, 0, stream>>>(batch, cnts, N);

    // final MLP -> out[64]
    mlp_kernel<<<1, B, 0, stream>>>(sums, cnts, Wm1, bm1, Wm2, bm2, (float*)d_out);
}